// GATNet_75265006895409
// MI455X (gfx1250) — compile-verified
//
#include <hip/hip_runtime.h>
#include <hip/hip_bf16.h>

// ---------------------------------------------------------------------------
// GATv2 network (BN -> GATv2(200->8x32) -> ReLU -> BN -> GATv2(32->8x16))
// MI455X / gfx1250, wave32. Node transforms use V_WMMA_F32_16X16X4_F32,
// 5 M-tiles per wave sharing each B fragment. Edge phase uses b128 loads
// + f32 L2 atomics for the segment softmax / aggregation.
// ---------------------------------------------------------------------------

#define N_NODES 50000
#define N_EDGES 400000
#define IN_DIM  200
#define HID     32
#define OUT_DIM 16
#define HEADS   8
#define NEG_SLOPE 0.2f
#define BN_EPS 1e-5f
#define MT      5            // M-tiles per wave in the WMMA GEMM (3125 = 625*5)

typedef __attribute__((ext_vector_type(2))) float v2f;
typedef __attribute__((ext_vector_type(8))) float v8f;

// ---- float <-> monotone-unsigned encoding for atomicMax on floats ---------
__device__ __forceinline__ unsigned enc_f(float x) {
    unsigned u = __float_as_uint(x);
    return (u & 0x80000000u) ? ~u : (u | 0x80000000u);
}
__device__ __forceinline__ float dec_f(unsigned u) {
    u = (u & 0x80000000u) ? (u & 0x7FFFFFFFu) : ~u;
    return __uint_as_float(u);
}

// ---------------------------------------------------------------------------
__global__ void fill_u32_kernel(unsigned* __restrict__ p, unsigned v, int n) {
    int i = blockIdx.x * blockDim.x + threadIdx.x;
    if (i < n) p[i] = v;
}

// Column partial sums for BatchNorm (coalesced: thread == column).
__global__ void bn_partial_kernel(const float* __restrict__ x,
                                  float* __restrict__ sum, float* __restrict__ sumsq,
                                  int rows, int cols, int rows_per_blk) {
    int c = threadIdx.x;
    if (c >= cols) return;
    int r0 = blockIdx.x * rows_per_blk;
    int r1 = min(r0 + rows_per_blk, rows);
    float s = 0.f, s2 = 0.f;
    for (int r = r0; r < r1; ++r) {
        float v = x[(size_t)r * cols + c];
        s += v; s2 += v * v;
    }
    atomicAdd(&sum[c], s);
    atomicAdd(&sumsq[c], s2);
}

__global__ void bn_finalize_kernel(const float* __restrict__ sum, const float* __restrict__ sumsq,
                                   float* __restrict__ mu, float* __restrict__ var,
                                   int cols, float inv_rows) {
    int c = blockIdx.x * blockDim.x + threadIdx.x;
    if (c < cols) {
        float m = sum[c] * inv_rows;
        mu[c] = m;
        var[c] = sumsq[c] * inv_rows - m * m;   // biased variance (matches jnp.var)
    }
}

__global__ void bn_apply_kernel(const float* __restrict__ x,
                                const float* __restrict__ mu, const float* __restrict__ var,
                                const float* __restrict__ gamma, const float* __restrict__ beta,
                                float* __restrict__ out, int total, int cols) {
    int i = blockIdx.x * blockDim.x + threadIdx.x;
    if (i >= total) return;
    int c = i % cols;
    out[i] = (x[i] - mu[c]) * rsqrtf(var[c] + BN_EPS) * gamma[c] + beta[c];
}

// ---------------------------------------------------------------------------
// C[M,N] = A[M,K] @ B[K,N], fp32, via V_WMMA_F32_16X16X4_F32.
// Each wave owns MT=5 vertically stacked 16x16 tiles (80 rows x 16 cols) and
// shares one B fragment per K-step across 5 WMMAs (independent accumulators,
// so no D->A/B RAW hazards -> full co-execution). 8 waves per block.
// M % (16*MT) == 0, K % 4 == 0, N % 16 == 0 (true for all calls here).
//
// A frag (16x4, 2 VGPRs/lane): lanes 0-15 row M=l hold K={k,k+1};
//   lanes 16-31 hold K={k+2,k+3}.
// B frag (4x16): VGPR0 = rows K={k, k+2} over lane halves, VGPR1 = {k+1,k+3}.
// C/D  : VGPR r = row (r + 8*half), col = lane&15.
// ---------------------------------------------------------------------------
__global__ void wmma_gemm_f32_kernel(const float* __restrict__ A, const float* __restrict__ B,
                                     float* __restrict__ C, int M, int K, int N) {
    int lane  = threadIdx.x & 31;
    int wave  = threadIdx.x >> 5;
    int tileM = blockIdx.x;                              // group of MT tiles
    int tileN = blockIdx.y * (blockDim.x >> 5) + wave;
    if (tileN * 16 >= N) return;                         // wave-uniform exit

    int l    = lane & 15;
    int half = lane >> 4;
    int n    = tileN * 16 + l;

    const float* Arow[MT];
#pragma unroll
    for (int t = 0; t < MT; ++t)
        Arow[t] = A + (size_t)((tileM * MT + t) * 16 + l) * K;

    v8f c[MT];
#pragma unroll
    for (int t = 0; t < MT; ++t) c[t] = (v8f){};

    for (int kk = 0; kk < K; kk += 4) {
        int ka = kk + 2 * half;
        v2f b;
        b[0] = B[(size_t)ka * N + n];
        b[1] = B[(size_t)(ka + 1) * N + n];
#pragma unroll
        for (int t = 0; t < MT; ++t) {
            v2f a;
            a[0] = Arow[t][ka];
            a[1] = Arow[t][ka + 1];
            c[t] = __builtin_amdgcn_wmma_f32_16x16x4_f32(false, a, false, b,
                                                         (short)0, c[t], false, false);
        }
    }

#pragma unroll
    for (int t = 0; t < MT; ++t) {
        int row0 = (tileM * MT + t) * 16 + half * 8;
#pragma unroll
        for (int r = 0; r < 8; ++r)
            C[(size_t)(row0 + r) * N + n] = c[t][r];
    }
}

// ---------------------------------------------------------------------------
// Edge phase. One thread per (edge, head); channel loops are float4-wide and
// fully unrolled (C is a template parameter). Self-loops: edges >= N_EDGES.
// ---------------------------------------------------------------------------
template <int C>
__global__ void edge_logits_kernel(const float* __restrict__ xl, const float* __restrict__ xr,
                                   const float* __restrict__ att, const int* __restrict__ ei,
                                   float* __restrict__ logits, unsigned* __restrict__ mmax) {
    int idx = blockIdx.x * blockDim.x + threadIdx.x;
    const int tot_e = N_EDGES + N_NODES;
    if (idx >= tot_e * HEADS) return;
    int e = idx / HEADS, hh = idx - e * HEADS;
    int s = (e < N_EDGES) ? ei[e]           : (e - N_EDGES);
    int d = (e < N_EDGES) ? ei[N_EDGES + e] : (e - N_EDGES);
    const float4* pl = (const float4*)(xl + ((size_t)s * HEADS + hh) * C);
    const float4* pr = (const float4*)(xr + ((size_t)d * HEADS + hh) * C);
    const float4* pa = (const float4*)(att + hh * C);
    float acc = 0.f;
#pragma unroll
    for (int c4 = 0; c4 < C / 4; ++c4) {
        float4 L = pl[c4], R = pr[c4], A = pa[c4];
        float v;
        v = L.x + R.x; v = (v > 0.f) ? v : NEG_SLOPE * v; acc += v * A.x;
        v = L.y + R.y; v = (v > 0.f) ? v : NEG_SLOPE * v; acc += v * A.y;
        v = L.z + R.z; v = (v > 0.f) ? v : NEG_SLOPE * v; acc += v * A.z;
        v = L.w + R.w; v = (v > 0.f) ? v : NEG_SLOPE * v; acc += v * A.w;
    }
    logits[idx] = acc;
    atomicMax(&mmax[d * HEADS + hh], enc_f(acc));
}

__global__ void edge_exp_kernel(float* __restrict__ logits, const unsigned* __restrict__ mmax,
                                float* __restrict__ denom, const int* __restrict__ ei) {
    int idx = blockIdx.x * blockDim.x + threadIdx.x;
    const int tot_e = N_EDGES + N_NODES;
    if (idx >= tot_e * HEADS) return;
    int e = idx / HEADS, hh = idx - e * HEADS;
    int d = (e < N_EDGES) ? ei[N_EDGES + e] : (e - N_EDGES);
    float ex = expf(logits[idx] - dec_f(mmax[d * HEADS + hh]));
    logits[idx] = ex;
    atomicAdd(&denom[d * HEADS + hh], ex);
}

template <int C>
__global__ void edge_scatter_kernel(const float* __restrict__ logits, const float* __restrict__ denom,
                                    const float* __restrict__ xl, float* __restrict__ acc,
                                    const int* __restrict__ ei) {
    int idx = blockIdx.x * blockDim.x + threadIdx.x;
    const int tot_e = N_EDGES + N_NODES;
    if (idx >= tot_e * HEADS) return;
    int e = idx / HEADS, hh = idx - e * HEADS;
    int s = (e < N_EDGES) ? ei[e]           : (e - N_EDGES);
    int d = (e < N_EDGES) ? ei[N_EDGES + e] : (e - N_EDGES);
    float alpha = logits[idx] / denom[d * HEADS + hh];
    const float4* pl = (const float4*)(xl + ((size_t)s * HEADS + hh) * C);
    float* pa = acc + ((size_t)d * HEADS + hh) * C;
#pragma unroll
    for (int c4 = 0; c4 < C / 4; ++c4) {
        float4 L = pl[c4];
        atomicAdd(&pa[4 * c4 + 0], L.x * alpha);
        atomicAdd(&pa[4 * c4 + 1], L.y * alpha);
        atomicAdd(&pa[4 * c4 + 2], L.z * alpha);
        atomicAdd(&pa[4 * c4 + 3], L.w * alpha);
    }
}

template <int C>
__global__ void head_mean_kernel(const float* __restrict__ acc, const float* __restrict__ bias,
                                 float* __restrict__ out, int do_relu) {
    int i = blockIdx.x * blockDim.x + threadIdx.x;
    if (i >= N_NODES * C) return;
    int n = i / C, c = i - n * C;
    float s = 0.f;
#pragma unroll
    for (int h = 0; h < HEADS; ++h) s += acc[((size_t)n * HEADS + h) * C + c];
    s = s * (1.0f / HEADS) + bias[c];
    if (do_relu) s = fmaxf(s, 0.f);
    out[i] = s;
}

// ---------------------------------------------------------------------------
extern "C" void kernel_launch(void* const* d_in, const int* in_sizes, int n_in,
                              void* d_out, int out_size, void* d_ws, size_t ws_size,
                              hipStream_t stream) {
    const float* x      = (const float*)d_in[0];
    const int*   ei     = (const int*)  d_in[1];   // [2, N_EDGES]
    // d_in[2] edge_weight: ignored (edge_dim is None in reference)
    const float* gamma1 = (const float*)d_in[3];
    const float* beta1  = (const float*)d_in[4];
    const float* Wl1    = (const float*)d_in[5];   // [200, 256]
    const float* Wr1    = (const float*)d_in[6];
    const float* att1   = (const float*)d_in[7];   // [8, 32]
    const float* b1     = (const float*)d_in[8];
    const float* gamma2 = (const float*)d_in[9];
    const float* beta2  = (const float*)d_in[10];
    const float* Wl2    = (const float*)d_in[11];  // [32, 128]
    const float* Wr2    = (const float*)d_in[12];
    const float* att2   = (const float*)d_in[13];  // [8, 16]
    const float* b2     = (const float*)d_in[14];
    float* out = (float*)d_out;

    // ---- workspace layout (floats) ----
    float* ws = (float*)d_ws;
    size_t off = 0;
    auto alloc = [&](size_t n) { float* p = ws + off; off += n; return p; };
    float*    h      = alloc((size_t)N_NODES * IN_DIM);          // BN1 output
    float*    xl     = alloc((size_t)N_NODES * HEADS * HID);     // reused layer2
    float*    xr     = alloc((size_t)N_NODES * HEADS * HID);
    float*    acc    = alloc((size_t)N_NODES * HEADS * HID);
    float*    logits = alloc((size_t)(N_EDGES + N_NODES) * HEADS);
    unsigned* mmax   = (unsigned*)alloc((size_t)N_NODES * HEADS);
    float*    denom  = alloc((size_t)N_NODES * HEADS);
    float*    bnsum  = alloc(256);
    float*    bnsq   = alloc(256);
    float*    mu     = alloc(256);
    float*    var    = alloc(256);
    float*    g1     = alloc((size_t)N_NODES * HID);             // relu(gat1)
    float*    h2     = alloc((size_t)N_NODES * HID);             // BN2 output

    const int TOT_E   = N_EDGES + N_NODES;
    const int EH      = TOT_E * HEADS;
    const int eh_blks = (EH + 255) / 256;
    const unsigned ENC_NEG_INF = 0x007FFFFFu;  // enc_f(-inf)

    // ============================ Layer 1 =================================
    // BN1
    fill_u32_kernel<<<2, 256, 0, stream>>>((unsigned*)bnsum, 0u, 512); // bnsum+bnsq
    bn_partial_kernel<<<(N_NODES + 511) / 512, 256, 0, stream>>>(x, bnsum, bnsq, N_NODES, IN_DIM, 512);
    bn_finalize_kernel<<<1, 256, 0, stream>>>(bnsum, bnsq, mu, var, IN_DIM, 1.0f / N_NODES);
    bn_apply_kernel<<<((N_NODES * IN_DIM) + 255) / 256, 256, 0, stream>>>(
        x, mu, var, gamma1, beta1, h, N_NODES * IN_DIM, IN_DIM);

    // xl = h @ Wl1, xr = h @ Wr1   [50000 x 256]
    {
        dim3 grid(N_NODES / (16 * MT), (HEADS * HID) / (16 * 8));
        wmma_gemm_f32_kernel<<<grid, 256, 0, stream>>>(h, Wl1, xl, N_NODES, IN_DIM, HEADS * HID);
        wmma_gemm_f32_kernel<<<grid, 256, 0, stream>>>(h, Wr1, xr, N_NODES, IN_DIM, HEADS * HID);
    }

    // segment softmax + aggregate
    fill_u32_kernel<<<(N_NODES * HEADS + 255) / 256, 256, 0, stream>>>(mmax, ENC_NEG_INF, N_NODES * HEADS);
    fill_u32_kernel<<<(N_NODES * HEADS + 255) / 256, 256, 0, stream>>>((unsigned*)denom, 0u, N_NODES * HEADS);
    fill_u32_kernel<<<(N_NODES * HEADS * HID + 255) / 256, 256, 0, stream>>>(
        (unsigned*)acc, 0u, N_NODES * HEADS * HID);

    edge_logits_kernel<HID><<<eh_blks, 256, 0, stream>>>(xl, xr, att1, ei, logits, mmax);
    edge_exp_kernel<<<eh_blks, 256, 0, stream>>>(logits, mmax, denom, ei);
    edge_scatter_kernel<HID><<<eh_blks, 256, 0, stream>>>(logits, denom, xl, acc, ei);
    head_mean_kernel<HID><<<(N_NODES * HID + 255) / 256, 256, 0, stream>>>(acc, b1, g1, 1);

    // ============================ Layer 2 =================================
    // BN2
    fill_u32_kernel<<<2, 256, 0, stream>>>((unsigned*)bnsum, 0u, 512);
    bn_partial_kernel<<<(N_NODES + 511) / 512, 256, 0, stream>>>(g1, bnsum, bnsq, N_NODES, HID, 512);
    bn_finalize_kernel<<<1, 256, 0, stream>>>(bnsum, bnsq, mu, var, HID, 1.0f / N_NODES);
    bn_apply_kernel<<<((N_NODES * HID) + 255) / 256, 256, 0, stream>>>(
        g1, mu, var, gamma2, beta2, h2, N_NODES * HID, HID);

    // xl = h2 @ Wl2, xr = h2 @ Wr2   [50000 x 128]
    {
        dim3 grid(N_NODES / (16 * MT), (HEADS * OUT_DIM) / (16 * 8));
        wmma_gemm_f32_kernel<<<grid, 256, 0, stream>>>(h2, Wl2, xl, N_NODES, HID, HEADS * OUT_DIM);
        wmma_gemm_f32_kernel<<<grid, 256, 0, stream>>>(h2, Wr2, xr, N_NODES, HID, HEADS * OUT_DIM);
    }

    fill_u32_kernel<<<(N_NODES * HEADS + 255) / 256, 256, 0, stream>>>(mmax, ENC_NEG_INF, N_NODES * HEADS);
    fill_u32_kernel<<<(N_NODES * HEADS + 255) / 256, 256, 0, stream>>>((unsigned*)denom, 0u, N_NODES * HEADS);
    fill_u32_kernel<<<(N_NODES * HEADS * OUT_DIM + 255) / 256, 256, 0, stream>>>(
        (unsigned*)acc, 0u, N_NODES * HEADS * OUT_DIM);

    edge_logits_kernel<OUT_DIM><<<eh_blks, 256, 0, stream>>>(xl, xr, att2, ei, logits, mmax);
    edge_exp_kernel<<<eh_blks, 256, 0, stream>>>(logits, mmax, denom, ei);
    edge_scatter_kernel<OUT_DIM><<<eh_blks, 256, 0, stream>>>(logits, denom, xl, acc, ei);
    head_mean_kernel<OUT_DIM><<<(N_NODES * OUT_DIM + 255) / 256, 256, 0, stream>>>(acc, b2, out, 0);
}